// SAGEConv2_76218489635041
// MI455X (gfx1250) — compile-verified
//
#include <hip/hip_runtime.h>
#include <hip/hip_bf16.h>

typedef __attribute__((ext_vector_type(2))) float v2f;
typedef __attribute__((ext_vector_type(8))) float v8f;

// ---------------------------------------------------------------------------
// Kernel 1: per-node projection  h[n] = dot(feat[n,:], sample_weights[:,col])
// One wave handles 2 nodes (16 lanes each); each lane loads a float4 of the
// 64-float row (16 lanes * 16B = 256B, fully coalesced). shfl_xor reduce
// with masks < 16 stays within each half-wave (wave32).
// mode==0 additionally writes coef[n] = norm_deg[n] / (q[n] * n_edges).
// ---------------------------------------------------------------------------
__global__ void proj_kernel(const float* __restrict__ feat,
                            const float* __restrict__ sw,   // [F,2] row-major
                            const float* __restrict__ norm_deg,
                            const float* __restrict__ q_probs,
                            float* __restrict__ h_out,
                            float* __restrict__ coef_out,
                            int n_nodes, int col, float inv_edges, int mode) {
    const int wave = (blockIdx.x * blockDim.x + threadIdx.x) >> 5;
    const int lane = threadIdx.x & 31;
    const int half = lane >> 4;          // which node within the wave
    const int l16  = lane & 15;
    const int node = wave * 2 + half;
    if (node >= n_nodes) return;

    const int f0 = l16 * 4;
    const float4 f = *(const float4*)(feat + (size_t)node * 64 + f0);
    const float w0 = sw[(f0 + 0) * 2 + col];
    const float w1 = sw[(f0 + 1) * 2 + col];
    const float w2 = sw[(f0 + 2) * 2 + col];
    const float w3 = sw[(f0 + 3) * 2 + col];
    float p = f.x * w0 + f.y * w1 + f.z * w2 + f.w * w3;

    // reduce across the 16-lane half-wave (masks 8,4,2,1 never cross halves)
    p += __shfl_xor(p, 8, 32);
    p += __shfl_xor(p, 4, 32);
    p += __shfl_xor(p, 2, 32);
    p += __shfl_xor(p, 1, 32);

    if (l16 == 0) {
        h_out[node] = p;
        if (mode == 0) {
            coef_out[node] = norm_deg[node] / q_probs[node] * inv_edges;
        }
    }
}

// ---------------------------------------------------------------------------
// Kernel 2: zero the neigh accumulator
// ---------------------------------------------------------------------------
__global__ void zero_kernel(float* __restrict__ p, int n) {
    int i = blockIdx.x * blockDim.x + threadIdx.x;
    if (i < n) p[i] = 0.0f;
}

// ---------------------------------------------------------------------------
// Kernel 3: per-edge attention scalar (coalesced, 1 thread per edge)
// att = coef_src[s] * norm_deg_dst[d] * (relu(hu[s]+hv[d]) + 0.1)
// (coef_src already contains norm_deg_src / q / n_edges)
// ---------------------------------------------------------------------------
__global__ void att_kernel(const int* __restrict__ src_idx,
                           const int* __restrict__ dst_idx,
                           const float* __restrict__ coef_src,
                           const float* __restrict__ ndd,
                           const float* __restrict__ hu,
                           const float* __restrict__ hv,
                           float* __restrict__ att, int n_edges) {
    int e = blockIdx.x * blockDim.x + threadIdx.x;
    if (e >= n_edges) return;
    int s = src_idx[e];
    int d = dst_idx[e];
    float g = fmaxf(hu[s] + hv[d], 0.0f) + 0.1f;
    att[e] = coef_src[s] * ndd[d] * g;
}

// ---------------------------------------------------------------------------
// Kernel 4: edge scatter. 16 lanes per edge; each lane gathers a float4 of
// hidden_feat[src] (256B per edge across the group) and issues 4 f32 global
// atomic adds into neigh[dst]. Non-returning -> global_atomic_add_f32.
// ---------------------------------------------------------------------------
__global__ void scatter_kernel(const int* __restrict__ src_idx,
                               const int* __restrict__ dst_idx,
                               const float* __restrict__ att,
                               const float* __restrict__ hidden,
                               float* __restrict__ neigh, int n_edges) {
    const int t = blockIdx.x * blockDim.x + threadIdx.x;
    const int e = t >> 4;
    if (e >= n_edges) return;
    const int f0 = (t & 15) * 4;
    const int s = src_idx[e];
    const int d = dst_idx[e];
    const float a = att[e];                      // 16 lanes hit same address -> broadcast
    const float4 h = *(const float4*)(hidden + (size_t)s * 64 + f0);
    float* base = neigh + (size_t)d * 64 + f0;
    atomicAdd(base + 0, h.x * a);
    atomicAdd(base + 1, h.y * a);
    atomicAdd(base + 2, h.z * a);
    atomicAdd(base + 3, h.w * a);
}

// ---------------------------------------------------------------------------
// Kernel 5: rst = neigh @ fc_weight^T + bias via V_WMMA_F32_16X16X4_F32.
// One wave per 16x16 output tile; K=64 in 16 chained WMMA steps.
// A (16x4 f32): lanes 0-15 hold M=lane with K=k,k+1 in v0,v1;
//               lanes 16-31 hold K=k+2,k+3 (per ISA layout).
// B (4x16 f32): symmetric; B[k][n] = fc_weight[n*64+k]  (W^T).
// C/D: VGPR i -> rows m0+i (lanes 0-15) / m0+8+i (lanes 16-31), col n0+l16.
// Tile guard is wave-uniform, so EXEC is all-1s for every executed WMMA.
// ---------------------------------------------------------------------------
__global__ void gemm_wmma_kernel(const float* __restrict__ neigh,
                                 const float* __restrict__ W,   // [64,64] row-major
                                 const float* __restrict__ bias,
                                 float* __restrict__ out,
                                 int n_tiles_total) {
    const int wave = (blockIdx.x * blockDim.x + threadIdx.x) >> 5;
    if (wave >= n_tiles_total) return;           // wave-uniform exit
    const int lane = threadIdx.x & 31;
    const int half = lane >> 4;
    const int l16  = lane & 15;

    const int mt = wave >> 2;                    // 4 N-tiles (F=64)
    const int nt = wave & 3;
    const int m0 = mt * 16;
    const int n0 = nt * 16;

    const float* arow = neigh + (size_t)(m0 + l16) * 64 + 2 * half;
    const float* brow = W     + (size_t)(n0 + l16) * 64 + 2 * half;

    v8f c = {};
#pragma unroll
    for (int k = 0; k < 64; k += 4) {
        v2f a = *(const v2f*)(arow + k);         // K = k+2*half, k+2*half+1
        v2f b = *(const v2f*)(brow + k);
        c = __builtin_amdgcn_wmma_f32_16x16x4_f32(
                false, a, false, b, (short)0, c, false, false);
    }

    const float bv = bias[n0 + l16];
#pragma unroll
    for (int i = 0; i < 8; ++i) {
        const int row = m0 + i + 8 * half;
        out[(size_t)row * 64 + n0 + l16] = c[i] + bv;
    }
}

// ---------------------------------------------------------------------------
// Host-side launch
// ---------------------------------------------------------------------------
extern "C" void kernel_launch(void* const* d_in, const int* in_sizes, int n_in,
                              void* d_out, int out_size, void* d_ws, size_t ws_size,
                              hipStream_t stream) {
    const float* hidden_feat   = (const float*)d_in[0];
    const float* node_feat_src = (const float*)d_in[1];
    const float* node_feat_dst = (const float*)d_in[2];
    const float* norm_deg_src  = (const float*)d_in[3];
    const float* norm_deg_dst  = (const float*)d_in[4];
    const float* q_probs       = (const float*)d_in[5];
    const float* sample_w      = (const float*)d_in[6];   // [F,2]
    const float* fc_weight     = (const float*)d_in[7];   // [F,F]
    const float* fc_bias       = (const float*)d_in[8];   // [F]
    const int*   src_idx       = (const int*)d_in[9];
    const int*   dst_idx       = (const int*)d_in[10];

    const int N_SRC = in_sizes[3];
    const int N_DST = in_sizes[4];
    const int E     = in_sizes[9];
    const int F     = 64;

    // workspace layout (floats)
    float* ws    = (float*)d_ws;
    float* neigh = ws;                         // N_DST * F
    float* hu    = neigh + (size_t)N_DST * F;  // N_SRC
    float* coef  = hu    + N_SRC;              // N_SRC
    float* hv    = coef  + N_SRC;              // N_DST
    float* att   = hv    + N_DST;              // E

    const float inv_edges = 1.0f / (float)E;

    // 1. projections (2 nodes per wave, 16 nodes per 256-thread block)
    {
        int blocks_s = (N_SRC + 15) / 16;
        proj_kernel<<<blocks_s, 256, 0, stream>>>(node_feat_src, sample_w,
                                                  norm_deg_src, q_probs,
                                                  hu, coef, N_SRC, 0, inv_edges, 0);
        int blocks_d = (N_DST + 15) / 16;
        proj_kernel<<<blocks_d, 256, 0, stream>>>(node_feat_dst, sample_w,
                                                  nullptr, nullptr,
                                                  hv, nullptr, N_DST, 1, inv_edges, 1);
    }

    // 2. zero the accumulator
    {
        int n = N_DST * F;
        zero_kernel<<<(n + 255) / 256, 256, 0, stream>>>(neigh, n);
    }

    // 3. per-edge attention
    att_kernel<<<(E + 255) / 256, 256, 0, stream>>>(src_idx, dst_idx, coef,
                                                    norm_deg_dst, hu, hv, att, E);

    // 4. scatter-accumulate (16 lanes per edge)
    {
        long long threads = (long long)E * 16;
        int blocks = (int)((threads + 255) / 256);
        scatter_kernel<<<blocks, 256, 0, stream>>>(src_idx, dst_idx, att,
                                                   hidden_feat, neigh, E);
    }

    // 5. output GEMM via WMMA (N_DST and F are multiples of 16)
    {
        int n_tiles = (N_DST / 16) * (F / 16);
        int waves_per_block = 256 / 32;
        int blocks = (n_tiles + waves_per_block - 1) / waves_per_block;
        gemm_wmma_kernel<<<blocks, 256, 0, stream>>>(neigh, fc_weight, fc_bias,
                                                     (float*)d_out, n_tiles);
    }
}